// GCNGraphRegression_7713761264261
// MI455X (gfx1250) — compile-verified
//
#include <hip/hip_runtime.h>

typedef float v2f __attribute__((ext_vector_type(2)));
typedef float v8f __attribute__((ext_vector_type(8)));

#define HD 128
#define EPSV 1e-5f

// ---------------- Embedding: x[n,c] = sum_j emb[xf[n,j] + 1 + j*512, c] ----
__global__ void embed_k(const int* __restrict__ xf, const float* __restrict__ emb,
                        float* __restrict__ x) {
    int node = blockIdx.x;
    int c = threadIdx.x;
    const int* f = xf + node * 9;
    float acc = 0.f;
#pragma unroll
    for (int j = 0; j < 9; ++j) {
        int idx = f[j] + 1 + j * 512;
        acc += emb[idx * HD + c];
    }
    x[node * HD + c] = acc;
}

// ---------------- utility fill ----------------
__global__ void fill_k(float* __restrict__ p, float v, int n) {
    int i = blockIdx.x * blockDim.x + threadIdx.x;
    if (i < n) p[i] = v;
}

// ---------------- degree (deg pre-filled with 1.0 for self-loops) ---------
__global__ void deg_k(const int* __restrict__ dst, float* __restrict__ deg, int E) {
    int i = blockIdx.x * blockDim.x + threadIdx.x;
    if (i < E) atomicAdd(&deg[dst[i]], 1.0f);
}
__global__ void dinv_k(float* __restrict__ deg, int n) {
    int i = blockIdx.x * blockDim.x + threadIdx.x;
    if (i < n) deg[i] = rsqrtf(deg[i]);   // deg >= 1 always (self-loop)
}

// ---------------- fp32 WMMA GEMM: C[M,128] = A[M,128] * B[128,128] --------
// one wave = one 16x16 tile; 8 waves/block cover all 128 columns.
// BIAS_RELU is compile-time so the store epilogue is branch-free.
template <bool BIAS_RELU>
__global__ void gemm_wmma_k(const float* __restrict__ A, const float* __restrict__ B,
                            float* __restrict__ C, const float* __restrict__ bias) {
    int wave = threadIdx.x >> 5;     // 0..7  -> column tile
    int lane = threadIdx.x & 31;
    int hi   = lane >> 4;            // 0 or 1 (K-half select)
    int l16  = lane & 15;
    int rb   = blockIdx.x * 16;      // row block
    int col  = wave * 16 + l16;      // output column for this lane

    const float* Arow = A + (rb + l16) * HD;
    v8f c = {0.f, 0.f, 0.f, 0.f, 0.f, 0.f, 0.f, 0.f};
#pragma unroll 4
    for (int k0 = 0; k0 < HD; k0 += 4) {
        v2f a, b;
        // A 16x4: lanes 0-15 -> {K=k0,k0+1}, lanes 16-31 -> {K=k0+2,k0+3}
        a.x = Arow[k0 + 2 * hi];
        a.y = Arow[k0 + 2 * hi + 1];
        // B 4x16: same K striping, N striped across lanes
        b.x = B[(k0 + 2 * hi) * HD + col];
        b.y = B[(k0 + 2 * hi + 1) * HD + col];
        c = __builtin_amdgcn_wmma_f32_16x16x4_f32(false, a, false, b,
                                                  (short)0, c, false, false);
    }
    float bv = BIAS_RELU ? bias[col] : 0.f;   // one load, hoisted
    // C layout: VGPR v -> row rb+v (lanes 0-15) / rb+v+8 (lanes 16-31)
#pragma unroll
    for (int v = 0; v < 8; ++v) {
        int row = rb + v + 8 * hi;
        float val = c[v];
        if (BIAS_RELU) val = fmaxf(val + bv, 0.f);
        C[row * HD + col] = val;
    }
}

// ---------------- agg init with per-layer bias ----------------
__global__ void agginit_k(float* __restrict__ agg, const float* __restrict__ bias) {
    agg[blockIdx.x * HD + threadIdx.x] = bias[threadIdx.x];
}

// ---------------- message passing: agg[dst] += h[src]*dinv[src]*dinv[dst] -
// edges e < E are real edges; e in [E, E+N) are self loops.
__global__ void scatter_k(const int* __restrict__ src, const int* __restrict__ dst,
                          const float* __restrict__ dinv, const float* __restrict__ h,
                          float* __restrict__ agg, int E, int n) {
    int e = blockIdx.x * 2 + (threadIdx.x >> 7);
    int c = threadIdx.x & 127;
    if (e >= E + n) return;
    int s, d;
    if (e < E) { s = src[e]; d = dst[e]; }
    else       { s = d = e - E; }
    float nrm = dinv[s] * dinv[d];
    atomicAdd(&agg[d * HD + c], h[s * HD + c] * nrm);
}

// ---------------- BN statistics (sum, sumsq per channel) ------------------
__global__ void bnstat_k(const float* __restrict__ agg, float* __restrict__ stats, int n) {
    int c = threadIdx.x;
    float s = 0.f, q = 0.f;
    for (int r = blockIdx.x; r < n; r += gridDim.x) {
        float v = agg[r * HD + c];
        s += v;
        q += v * v;
    }
    atomicAdd(&stats[c], s);
    atomicAdd(&stats[HD + c], q);
}

// ---------------- BN apply + ReLU ----------------
__global__ void bnrelu_k(const float* __restrict__ agg, const float* __restrict__ stats,
                         const float* __restrict__ g, const float* __restrict__ bt,
                         float* __restrict__ x, int n) {
    int node = blockIdx.x;
    int c = threadIdx.x;
    float invn = 1.0f / (float)n;
    float m = stats[c] * invn;
    float var = stats[HD + c] * invn - m * m;
    float sc = g[c] * rsqrtf(var + EPSV);
    float v = (agg[node * HD + c] - m) * sc + bt[c];
    x[node * HD + c] = fmaxf(v, 0.f);
}

// ---------------- global mean pool ----------------
__global__ void pool_k(const float* __restrict__ x, const int* __restrict__ batch,
                       float* __restrict__ pooled, float* __restrict__ counts) {
    int node = blockIdx.x;
    int c = threadIdx.x;
    int g = batch[node];
    atomicAdd(&pooled[g * HD + c], x[node * HD + c]);
    if (c == 0) atomicAdd(&counts[g], 1.0f);
}
__global__ void poolavg_k(float* __restrict__ pooled, const float* __restrict__ counts) {
    int g = blockIdx.x;
    pooled[g * HD + threadIdx.x] /= fmaxf(counts[g], 1.0f);
}

// ---------------- final BN over G rows (small; one block) -----------------
__global__ void bnfinal_k(float* __restrict__ pooled, const float* __restrict__ gm,
                          const float* __restrict__ bt, int G) {
    int c = threadIdx.x;
    float s = 0.f, q = 0.f;
    for (int g = 0; g < G; ++g) {
        float v = pooled[g * HD + c];
        s += v; q += v * v;
    }
    float m = s / (float)G;
    float var = q / (float)G - m * m;
    float sc = gm[c] * rsqrtf(var + EPSV);
    for (int g = 0; g < G; ++g)
        pooled[g * HD + c] = (pooled[g * HD + c] - m) * sc + bt[c];
}

// ---------------- lin2: out[g] = dot(hmid[g,:], w) + b ----------------
__global__ void lin2_k(const float* __restrict__ hmid, const float* __restrict__ w,
                       const float* __restrict__ b2, float* __restrict__ out) {
    __shared__ float red[HD];
    int g = blockIdx.x;
    int t = threadIdx.x;
    red[t] = hmid[g * HD + t] * w[t];
    __syncthreads();
    for (int s = 64; s > 0; s >>= 1) {
        if (t < s) red[t] += red[t + s];
        __syncthreads();
    }
    if (t == 0) out[g] = red[0] + b2[0];
}

extern "C" void kernel_launch(void* const* d_in, const int* in_sizes, int n_in,
                              void* d_out, int out_size, void* d_ws, size_t ws_size,
                              hipStream_t stream) {
    const int*   x_feat   = (const int*)d_in[0];     // [N,9]
    const int*   eidx     = (const int*)d_in[1];     // [2,E]
    const int*   batch    = (const int*)d_in[2];     // [N]
    const float* emb      = (const float*)d_in[3];   // [VOCAB,128]
    const float* W        = (const float*)d_in[4];   // [L,128,128]
    const float* bvec     = (const float*)d_in[5];   // [L,128]
    const float* gamma    = (const float*)d_in[6];   // [L,128]
    const float* beta     = (const float*)d_in[7];   // [L,128]
    const float* gamma_f  = (const float*)d_in[8];   // [128]
    const float* beta_f   = (const float*)d_in[9];   // [128]
    const float* lin1_w   = (const float*)d_in[10];  // [128,128]
    const float* lin1_b   = (const float*)d_in[11];  // [128]
    const float* lin2_w   = (const float*)d_in[12];  // [128]
    const float* lin2_b   = (const float*)d_in[13];  // [1]
    float* out = (float*)d_out;

    const int N = in_sizes[0] / 9;
    const int E = in_sizes[1] / 2;
    const int G = out_size;
    const int L = 4;

    const int* src = eidx;
    const int* dst = eidx + E;

    // workspace layout (floats)
    float* ws     = (float*)d_ws;
    float* x      = ws;                     // N*128
    float* h      = x + (size_t)N * HD;     // N*128
    float* agg    = h + (size_t)N * HD;     // N*128
    float* dinv   = agg + (size_t)N * HD;   // N
    float* stats  = dinv + N;               // 2*128
    float* pooled = stats + 2 * HD;         // G*128
    float* counts = pooled + (size_t)G * HD;// G
    float* hmid   = counts + G;             // G*128

    // --- embedding ---
    embed_k<<<N, HD, 0, stream>>>(x_feat, emb, x);

    // --- degree / normalization ---
    fill_k<<<(N + 255) / 256, 256, 0, stream>>>(dinv, 1.0f, N);   // self-loops
    deg_k<<<(E + 255) / 256, 256, 0, stream>>>(dst, dinv, E);
    dinv_k<<<(N + 255) / 256, 256, 0, stream>>>(dinv, N);

    // --- GCN layers ---
    for (int i = 0; i < L; ++i) {
        gemm_wmma_k<false><<<N / 16, 256, 0, stream>>>(x, W + (size_t)i * HD * HD, h,
                                                       nullptr);
        agginit_k<<<N, HD, 0, stream>>>(agg, bvec + i * HD);
        scatter_k<<<(E + N + 1) / 2, 256, 0, stream>>>(src, dst, dinv, h, agg, E, N);
        fill_k<<<1, 256, 0, stream>>>(stats, 0.0f, 2 * HD);
        bnstat_k<<<512, HD, 0, stream>>>(agg, stats, N);
        bnrelu_k<<<N, HD, 0, stream>>>(agg, stats, gamma + i * HD, beta + i * HD, x, N);
    }

    // --- global mean pool ---
    fill_k<<<(G * HD + G + 255) / 256, 256, 0, stream>>>(pooled, 0.0f, G * HD + G);
    pool_k<<<N, HD, 0, stream>>>(x, batch, pooled, counts);
    poolavg_k<<<G, HD, 0, stream>>>(pooled, counts);

    // --- final BN + MLP head ---
    bnfinal_k<<<1, HD, 0, stream>>>(pooled, gamma_f, beta_f, G);
    gemm_wmma_k<true><<<G / 16, 256, 0, stream>>>(pooled, lin1_w, hmid, lin1_b);
    lin2_k<<<G, HD, 0, stream>>>(hmid, lin2_w, lin2_b, out);
}